// MS_Loss_65841848647901
// MI455X (gfx1250) — compile-verified
//
#include <hip/hip_runtime.h>
#include <math.h>

// Problem constants (match reference)
constexpr int kB = 8, kT = 1024, kF = 1024, kR = 4, kDK = 64, kNB = 2;
constexpr float kEPS = 1e-5f;

typedef __attribute__((ext_vector_type(2))) float v2f;
typedef __attribute__((ext_vector_type(8))) float v8f;

// CDNA5 fp32 matrix core: D = A(16x4) * B(4x16) + C(16x16), full fp32.
__device__ __forceinline__ v8f wmma4(v2f a, v2f b, v8f c) {
  return __builtin_amdgcn_wmma_f32_16x16x4_f32(
      /*neg_a=*/false, a, /*neg_b=*/false, b,
      /*c_mod=*/(short)0, c, /*reuse_a=*/false, /*reuse_b=*/false);
}

// XOR swizzle for the 16x1024 LDS S-buffer: makes A-fragment (column-wise
// across rows) reads conflict-free across the 64 LDS banks.
__device__ __forceinline__ int sidx(int row, int s) {
  return row * 1024 + (s ^ ((row & 15) << 2));
}

// ---------------------------------------------------------------------------
// LayerNorm over (T,F) jointly, per batch. 3-stage for parallelism.
// ---------------------------------------------------------------------------
__global__ void ln_partial_kernel(const float* __restrict__ src,
                                  float* __restrict__ partial) {
  const int chunk = blockIdx.x;  // 0..63
  const int b = blockIdx.y;
  const int per_chunk = kT * kF / 64;  // 16384
  const size_t base = (size_t)b * kT * kF + (size_t)chunk * per_chunk;
  float s1 = 0.f, s2 = 0.f;
  for (int i = threadIdx.x; i < per_chunk; i += 256) {
    float v = src[base + i];
    s1 += v;
    s2 += v * v;
  }
  __shared__ float r1[256], r2[256];
  r1[threadIdx.x] = s1;
  r2[threadIdx.x] = s2;
  __syncthreads();
  for (int off = 128; off; off >>= 1) {
    if ((int)threadIdx.x < off) {
      r1[threadIdx.x] += r1[threadIdx.x + off];
      r2[threadIdx.x] += r2[threadIdx.x + off];
    }
    __syncthreads();
  }
  if (threadIdx.x == 0) {
    partial[(b * 64 + chunk) * 2 + 0] = r1[0];
    partial[(b * 64 + chunk) * 2 + 1] = r2[0];
  }
}

__global__ void ln_finalize_kernel(const float* __restrict__ partial,
                                   float* __restrict__ stats) {
  const int b = blockIdx.x;
  __shared__ float r1[64], r2[64];
  const int t = threadIdx.x;
  r1[t] = partial[(b * 64 + t) * 2 + 0];
  r2[t] = partial[(b * 64 + t) * 2 + 1];
  __syncthreads();
  for (int off = 32; off; off >>= 1) {
    if (t < off) {
      r1[t] += r1[t + off];
      r2[t] += r2[t + off];
    }
    __syncthreads();
  }
  if (t == 0) {
    const float n = (float)(kT * kF);
    float mean = r1[0] / n;
    float var = r2[0] / n - mean * mean;
    stats[b * 2 + 0] = mean;
    stats[b * 2 + 1] = rsqrtf(var + kEPS);
  }
}

__global__ void ln_apply_kernel(const float* __restrict__ src,
                                const float* __restrict__ resid,
                                const float* __restrict__ w,
                                const float* __restrict__ bia,
                                const float* __restrict__ stats,
                                float* __restrict__ dst) {
  const size_t i = ((size_t)blockIdx.x * 256 + threadIdx.x) * 4;
  const size_t per = (size_t)kT * kF;
  const int b = (int)(i / per);
  const float mean = stats[b * 2 + 0];
  const float inv = stats[b * 2 + 1];
  const size_t j = i - (size_t)b * per;  // index inside (T,F) for w/b
  float4 v = *(const float4*)(src + i);
  float4 wv = *(const float4*)(w + j);
  float4 bv = *(const float4*)(bia + j);
  float4 o;
  o.x = (v.x - mean) * inv * wv.x + bv.x;
  o.y = (v.y - mean) * inv * wv.y + bv.y;
  o.z = (v.z - mean) * inv * wv.z + bv.z;
  o.w = (v.w - mean) * inv * wv.w + bv.w;
  if (resid) {
    float4 rv = *(const float4*)(resid + i);
    o.x += rv.x; o.y += rv.y; o.z += rv.z; o.w += rv.w;
  }
  *(float4*)(dst + i) = o;
}

// ---------------------------------------------------------------------------
// Q/K projection: out[b,r,t,:] = h[b,t,:] @ W[r] + bias[r]
// Grid (T/16, R, B), 128 threads (4 waves; wave w -> DK tile w).
// ---------------------------------------------------------------------------
__global__ void qk_proj_kernel(const float* __restrict__ h,
                               const float* __restrict__ W,     // [R][F][DK]
                               const float* __restrict__ bias,  // [R][DK]
                               float* __restrict__ out) {       // [B][R][T][DK]
  const int ttile = blockIdx.x;
  const int r = blockIdx.y;
  const int b = blockIdx.z;
  const int wave = threadIdx.x >> 5;
  const int lane = threadIdx.x & 31;
  const int m = lane & 15;
  const int hl = lane >> 4;  // 0/1: K-halves of the A/B fragments
  const float* hB = h + (size_t)b * kT * kF + (size_t)(ttile * 16 + m) * kF;
  const float* Wr = W + (size_t)r * kF * kDK;
  const int n = wave * 16 + m;  // DK column
  v8f acc = {};
  for (int k0 = 0; k0 < kF; k0 += 4) {
    const int ka = k0 + hl * 2;
    v2f a;
    a.x = hB[ka];
    a.y = hB[ka + 1];
    v2f bb;
    bb.x = Wr[(size_t)ka * kDK + n];
    bb.y = Wr[(size_t)(ka + 1) * kDK + n];
    acc = wmma4(a, bb, acc);
  }
  const float bval = bias[r * kDK + n];
  float* o = out + (((size_t)b * kR + r) * kT + ttile * 16) * kDK + n;
#pragma unroll
  for (int i = 0; i < 8; ++i) {
    o[(size_t)(i + hl * 8) * kDK] = acc[i] + bval;
  }
}

// ---------------------------------------------------------------------------
// Fused attention + head-sum: mid[b,t,:] = sum_r softmax(Q Kt / 8) @ h[b]
// Grid (T/16, B), 256 threads (8 waves). Per head: S-rowblock in LDS,
// shuffle softmax, then each wave accumulates a 128-wide F stripe in regs.
// ---------------------------------------------------------------------------
__global__ void attn_kernel(const float* __restrict__ Q,  // [B][R][T][DK]
                            const float* __restrict__ K,  // [B][R][T][DK]
                            const float* __restrict__ h,  // [B][T][F]
                            float* __restrict__ mid) {    // [B][T][F]
  __shared__ float Sbuf[16 * 1024];  // 64 KB, XOR-swizzled
  const int ttile = blockIdx.x;
  const int b = blockIdx.y;
  const int wave = threadIdx.x >> 5;
  const int lane = threadIdx.x & 31;
  const int m = lane & 15;
  const int hl = lane >> 4;
  const float scale = 0.125f;  // 1/sqrt(DK)

  v8f acc[8];
  {
    v8f z = {};
#pragma unroll
    for (int i = 0; i < 8; ++i) acc[i] = z;
  }
  const float* hb = h + (size_t)b * kT * kF;

  for (int r = 0; r < kR; ++r) {
    const float* Qb = Q + (((size_t)b * kR + r) * kT + ttile * 16) * kDK;
    const float* Kb = K + (((size_t)b * kR + r) * kT) * kDK;

    // Preload this tile's Q A-fragments (16 K-steps over DK=64)
    v2f qa[16];
#pragma unroll
    for (int j = 0; j < 16; ++j) {
      const int ka = 4 * j + hl * 2;
      qa[j].x = Qb[(size_t)m * kDK + ka];
      qa[j].y = Qb[(size_t)m * kDK + ka + 1];
    }

    // Phase 1: S[16][1024] = Q_tile @ K^T * scale  (8 s-tiles per wave)
    for (int nt = wave; nt < 64; nt += 8) {
      v8f c = {};
      const float* Krow = Kb + (size_t)(nt * 16 + m) * kDK;
#pragma unroll
      for (int j = 0; j < 16; ++j) {
        const int ka = 4 * j + hl * 2;
        v2f bb;
        bb.x = Krow[ka];
        bb.y = Krow[ka + 1];
        c = wmma4(qa[j], bb, c);
      }
#pragma unroll
      for (int i = 0; i < 8; ++i) {
        const int mr = i + hl * 8;
        Sbuf[sidx(mr, nt * 16 + m)] = c[i] * scale;
      }
    }
    __syncthreads();

    // Phase 2: row softmax (wave handles rows 2*wave, 2*wave+1)
    for (int rr = 0; rr < 2; ++rr) {
      const int row = wave * 2 + rr;
      float mx = -3.402823466e38f;
      for (int s = lane; s < 1024; s += 32)
        mx = fmaxf(mx, Sbuf[sidx(row, s)]);
      for (int off = 16; off; off >>= 1) mx = fmaxf(mx, __shfl_xor(mx, off, 32));
      float sum = 0.f;
      for (int s = lane; s < 1024; s += 32) {
        const int idx = sidx(row, s);
        const float e = __expf(Sbuf[idx] - mx);
        Sbuf[idx] = e;
        sum += e;
      }
      for (int off = 16; off; off >>= 1) sum += __shfl_xor(sum, off, 32);
      const float invs = 1.0f / sum;
      for (int s = lane; s < 1024; s += 32) Sbuf[sidx(row, s)] *= invs;
    }
    __syncthreads();

    // Phase 3: acc += P @ h, wave's 128-col F stripe (8 tiles, K=1024)
    for (int ft = 0; ft < 8; ++ft) {
      const int fcol = wave * 128 + ft * 16 + m;
      v8f c = acc[ft];
      for (int k0 = 0; k0 < 1024; k0 += 4) {
        const int ka = k0 + hl * 2;
        v2f a;
        a.x = Sbuf[sidx(m, ka)];
        a.y = Sbuf[sidx(m, ka + 1)];
        v2f bb;
        bb.x = hb[(size_t)ka * kF + fcol];
        bb.y = hb[(size_t)(ka + 1) * kF + fcol];
        c = wmma4(a, bb, c);
      }
      acc[ft] = c;
    }
    __syncthreads();  // before next head rewrites Sbuf
  }

  // Store mid (head-summed attention output)
  float* mo = mid + ((size_t)b * kT + ttile * 16) * kF;
#pragma unroll
  for (int ft = 0; ft < 8; ++ft) {
    const int fcol = wave * 128 + ft * 16 + m;
#pragma unroll
    for (int i = 0; i < 8; ++i) {
      mo[(size_t)(i + hl * 8) * kF + fcol] = acc[ft][i];
    }
  }
}

// ---------------------------------------------------------------------------
// out[b,t] = R * (h[b,t,:] . Wp[:,0] + bp[0]) ; one wave per row.
// ---------------------------------------------------------------------------
__global__ void proj_out_kernel(const float* __restrict__ h,
                                const float* __restrict__ Wp,
                                const float* __restrict__ bp,
                                float* __restrict__ out) {
  const int wave = threadIdx.x >> 5;
  const int lane = threadIdx.x & 31;
  const int row = blockIdx.x * 8 + wave;  // 0..B*T-1
  const float* hr = h + (size_t)row * kF;
  float s = 0.f;
  for (int f = lane; f < kF; f += 32) s += hr[f] * Wp[f];
  for (int off = 16; off; off >>= 1) s += __shfl_xor(s, off, 32);
  if (lane == 0) out[row] = (float)kR * (s + bp[0]);
}

// ---------------------------------------------------------------------------
extern "C" void kernel_launch(void* const* d_in, const int* in_sizes, int n_in,
                              void* d_out, int out_size, void* d_ws,
                              size_t ws_size, hipStream_t stream) {
  (void)in_sizes; (void)n_in; (void)out_size; (void)ws_size;
  const float* x = (const float*)d_in[0];
  const float* Wq = (const float*)d_in[1];
  const float* bq = (const float*)d_in[2];
  const float* Wk = (const float*)d_in[3];
  const float* bk = (const float*)d_in[4];
  const float* Wp = (const float*)d_in[5];
  const float* bp = (const float*)d_in[6];
  const float* lnw = (const float*)d_in[7];
  const float* lnb = (const float*)d_in[8];

  float* ws = (float*)d_ws;
  float* h = ws;                                   // B*T*F
  float* mid = h + (size_t)kB * kT * kF;           // B*T*F
  float* Qb = mid + (size_t)kB * kT * kF;          // B*R*T*DK
  float* Kb = Qb + (size_t)kB * kR * kT * kDK;     // B*R*T*DK
  float* partial = Kb + (size_t)kB * kR * kT * kDK;  // B*64*2
  float* stats = partial + kB * 64 * 2;            // B*2

  // LN0: x -> h
  ln_partial_kernel<<<dim3(64, kB), 256, 0, stream>>>(x, partial);
  ln_finalize_kernel<<<kB, 64, 0, stream>>>(partial, stats);
  ln_apply_kernel<<<(kB * kT * kF) / 1024, 256, 0, stream>>>(
      x, nullptr, lnw, lnb, stats, h);

  for (int blk = 0; blk < kNB; ++blk) {
    const float* Wqb = Wq + (size_t)blk * kR * kF * kDK;
    const float* bqb = bq + (size_t)blk * kR * kDK;
    const float* Wkb = Wk + (size_t)blk * kR * kF * kDK;
    const float* bkb = bk + (size_t)blk * kR * kDK;

    qk_proj_kernel<<<dim3(kT / 16, kR, kB), 128, 0, stream>>>(h, Wqb, bqb, Qb);
    qk_proj_kernel<<<dim3(kT / 16, kR, kB), 128, 0, stream>>>(h, Wkb, bkb, Kb);
    attn_kernel<<<dim3(kT / 16, kB), 256, 0, stream>>>(Qb, Kb, h, mid);

    ln_partial_kernel<<<dim3(64, kB), 256, 0, stream>>>(mid, partial);
    ln_finalize_kernel<<<kB, 64, 0, stream>>>(partial, stats);
    ln_apply_kernel<<<(kB * kT * kF) / 1024, 256, 0, stream>>>(
        mid, h, lnw + (size_t)(blk + 1) * kT * kF,
        lnb + (size_t)(blk + 1) * kT * kF, stats, h);
  }

  proj_out_kernel<<<(kB * kT) / 8, 256, 0, stream>>>(h, Wp, bp, (float*)d_out);
}